// InternVLWithExpertModel_59717225284170
// MI455X (gfx1250) — compile-verified
//
#include <hip/hip_runtime.h>
#include <math.h>

// ---------------- problem constants (from reference) ----------------
#define H_    32
#define KVH_  8
#define DH_   128
#define G_    4            // H / KVH
#define B_    2
#define LV_   1024
#define LE_   64
#define L_    (LV_ + LE_)  // 1088
#define DV_   2560
#define DE_   1920
#define QDIM  (H_ * DH_)   // 4096
#define KVDIM (KVH_ * DH_) // 1024

typedef __attribute__((ext_vector_type(2))) float v2f;
typedef __attribute__((ext_vector_type(8))) float v8f;

// V_WMMA_F32_16X16X4_F32 : D(16x16,f32) = A(16x4,f32) * B(4x16,f32) + C
// A frag (2 VGPRs): lane r=lane&15 -> row M=r ; VGPR v, group g=lane>>4 -> K = v + 2g
// B frag (2 VGPRs): lane r -> col N=r ; VGPR v, group g -> K = v + 2g
// C/D  (8 VGPRs):   lane r -> col N=r ; VGPR v, group g -> row M = v + 8g
__device__ __forceinline__ v8f wmma4_f32(v2f a, v2f b, v8f c) {
  return __builtin_amdgcn_wmma_f32_16x16x4_f32(
      /*neg_a=*/false, a, /*neg_b=*/false, b,
      /*c_mod=*/(short)0, c, /*reuse_a=*/false, /*reuse_b=*/false);
}

// async global -> LDS (ASYNCcnt); per-lane 16B
__device__ __forceinline__ void async_ld_b128(unsigned lds_byte_off,
                                              unsigned long long gaddr) {
  asm volatile("global_load_async_to_lds_b128 %0, %1, off"
               :: "v"(lds_byte_off), "v"(gaddr) : "memory");
}
__device__ __forceinline__ unsigned lds_off(const void* p) {
  return (unsigned)(unsigned long long)p;  // low 32 bits of generic = LDS offset
}

// ============ tiled GEMM with async double buffering =================
// C[M,N] = A[M,K] @ W[K,N], fp32 row-major.
// block tile 128(M) x 64(N), K-slice 32, 256 threads = 8 waves (4Mx2N),
// each wave computes 2x2 WMMA 16x16 tiles (32x32 region).
#define BM   128
#define BN   64
#define BK   32
#define ALD  (BK + 2)          // padded A row stride in LDS (floats)
#define SA_SZ (BM * ALD)
#define SB_SZ (BK * BN)

__global__ void gemm_tiled_async(const float* __restrict__ A,
                                 const float* __restrict__ W,
                                 float* __restrict__ C,
                                 int M, int N, int K) {
  __shared__ float sA[2][SA_SZ];
  __shared__ float sB[2][SB_SZ];

  const int t = threadIdx.x;
  const int blockN = blockIdx.x * BN;
  const int blockM = blockIdx.y * BM;

  // issue one block tile's async loads (6 wave-instructions per wave)
  auto load_tile = [&](int buf, int kc) {
    const unsigned aBase = lds_off(&sA[buf][0]);
    const unsigned bBase = lds_off(&sB[buf][0]);
#pragma unroll
    for (int i = 0; i < 4; ++i) {              // A: 128 x 32, 4-float granules
      const int idx = t + 256 * i;
      const int row = idx >> 3;                // 0..127
      const int cg  = idx & 7;                 // 0..7
      const unsigned long long ga =
          (unsigned long long)(A + (size_t)(blockM + row) * K + kc + cg * 4);
      async_ld_b128(aBase + (unsigned)(row * ALD + cg * 4) * 4u, ga);
    }
#pragma unroll
    for (int i = 0; i < 2; ++i) {              // B: 32 x 64
      const int idx = t + 256 * i;
      const int row = idx >> 4;                // 0..31
      const int cg  = idx & 15;                // 0..15
      const unsigned long long ga =
          (unsigned long long)(W + (size_t)(kc + row) * N + blockN + cg * 4);
      async_ld_b128(bBase + (unsigned)(row * BN + cg * 4) * 4u, ga);
    }
  };

  const int lane = t & 31;
  const int wave = t >> 5;
  const int wm = wave >> 1;      // 0..3 -> M offset wm*32
  const int wn = wave & 1;       // 0..1 -> N offset wn*32
  const int r = lane & 15;
  const int g = lane >> 4;

  v8f acc00 = {}, acc01 = {}, acc10 = {}, acc11 = {};

  int buf = 0;
  load_tile(0, 0);
  for (int kc = 0; kc < K; kc += BK) {
    if (kc + BK < K) {
      load_tile(buf ^ 1, kc + BK);
      asm volatile("s_wait_asynccnt 0x6" ::: "memory"); // oldest 6 (this buf) done
    } else {
      asm volatile("s_wait_asynccnt 0x0" ::: "memory");
    }
    __syncthreads();

    const float* a0 = &sA[buf][(wm * 32 + r) * ALD];
    const float* a1 = &sA[buf][(wm * 32 + 16 + r) * ALD];
    const float* bp = &sB[buf][wn * 32 + r];
#pragma unroll
    for (int k = 0; k < BK; k += 4) {
      v2f af0 = *(const v2f*)(a0 + k + 2 * g);
      v2f af1 = *(const v2f*)(a1 + k + 2 * g);
      v2f bf0, bf1;
      bf0.x = bp[(k + 2 * g + 0) * BN];
      bf0.y = bp[(k + 2 * g + 1) * BN];
      bf1.x = bp[(k + 2 * g + 0) * BN + 16];
      bf1.y = bp[(k + 2 * g + 1) * BN + 16];
      acc00 = wmma4_f32(af0, bf0, acc00);
      acc01 = wmma4_f32(af0, bf1, acc01);
      acc10 = wmma4_f32(af1, bf0, acc10);
      acc11 = wmma4_f32(af1, bf1, acc11);
    }
    __syncthreads();   // all waves done reading buf before it is reloaded
    buf ^= 1;
  }

  float* Cbase = C + (size_t)(blockM + wm * 32) * N + blockN + wn * 32 + r;
#pragma unroll
  for (int v = 0; v < 8; ++v) {
    const int row0 = v + 8 * g;
    Cbase[(size_t)row0 * N]            = acc00[v];
    Cbase[(size_t)row0 * N + 16]       = acc01[v];
    Cbase[(size_t)(16 + row0) * N]     = acc10[v];
    Cbase[(size_t)(16 + row0) * N + 16] = acc11[v];
  }
}

// ---------------- generic GEMM: C[M,N] = A[M,K] @ W[K,N] (small M cases)
__global__ void gemm_wmma_f32(const float* __restrict__ A,
                              const float* __restrict__ W,
                              float* __restrict__ C,
                              int M, int N, int K) {
  const int lane = threadIdx.x & 31;
  const int tile = blockIdx.x * (blockDim.x >> 5) + (threadIdx.x >> 5);
  const int ntile = N >> 4;
  const int tm = tile / ntile;
  const int tn = tile - tm * ntile;
  if (tm >= (M >> 4)) return;
  const int r = lane & 15;
  const int g = lane >> 4;
  const float* Arow = A + (size_t)(tm * 16 + r) * K;
  const float* Wcol = W + (size_t)(tn * 16 + r);
  v8f acc = {};
  for (int k0 = 0; k0 < K; k0 += 4) {
    v2f a = *(const v2f*)(Arow + k0 + 2 * g);
    v2f b;
    b.x = Wcol[(size_t)(k0 + 2 * g + 0) * N];
    b.y = Wcol[(size_t)(k0 + 2 * g + 1) * N];
    acc = wmma4_f32(a, b, acc);
  }
#pragma unroll
  for (int v = 0; v < 8; ++v)
    C[(size_t)(tm * 16 + v + 8 * g) * N + tn * 16 + r] = acc[v];
}

// ---------------- out_proj: out = res + O[M,K] @ W[K,N] --------------
__global__ void out_proj_wmma(const float* __restrict__ O,
                              const float* __restrict__ W,
                              const float* __restrict__ res,
                              float* __restrict__ out,
                              int M, int N, int K) {
  const int lane = threadIdx.x & 31;
  const int tile = blockIdx.x * (blockDim.x >> 5) + (threadIdx.x >> 5);
  const int ntile = N >> 4;
  const int tm = tile / ntile;
  const int tn = tile - tm * ntile;
  if (tm >= (M >> 4)) return;
  const int r = lane & 15;
  const int g = lane >> 4;
  const float* Arow = O + (size_t)(tm * 16 + r) * K;
  const float* Wcol = W + (size_t)(tn * 16 + r);
  v8f acc = {};
  for (int k0 = 0; k0 < K; k0 += 4) {
    v2f a = *(const v2f*)(Arow + k0 + 2 * g);
    v2f b;
    b.x = Wcol[(size_t)(k0 + 2 * g + 0) * N];
    b.y = Wcol[(size_t)(k0 + 2 * g + 1) * N];
    acc = wmma4_f32(a, b, acc);
  }
#pragma unroll
  for (int v = 0; v < 8; ++v) {
    size_t idx = (size_t)(tm * 16 + v + 8 * g) * N + tn * 16 + r;
    out[idx] = res[idx] + acc[v];
  }
}

// ---------------- RMS norm: one 256-thread block per row -------------
__global__ void rmsnorm_kernel(const float* __restrict__ X,
                               const float* __restrict__ w,
                               float* __restrict__ Y, int D) {
  const int row = blockIdx.x;
  const float* x = X + (size_t)row * D;
  float* y = Y + (size_t)row * D;
  __shared__ float red[8];
  __shared__ float stot;
  float ss = 0.f;
  for (int c = threadIdx.x; c < D; c += blockDim.x) {
    float v = x[c];
    ss += v * v;
  }
  for (int off = 16; off > 0; off >>= 1) ss += __shfl_xor(ss, off, 32);
  if ((threadIdx.x & 31) == 0) red[threadIdx.x >> 5] = ss;
  __syncthreads();
  if (threadIdx.x == 0) {
    float tsum = 0.f;
    for (int i = 0; i < (int)(blockDim.x >> 5); ++i) tsum += red[i];
    stot = tsum;
  }
  __syncthreads();
  const float inv = rsqrtf(stot / (float)D + 1e-6f);
  for (int c = threadIdx.x; c < D; c += blockDim.x) y[c] = x[c] * inv * w[c];
}

// ---------------- RoPE (in place), d_half = 64 -----------------------
__global__ void rope_k_kernel(float* __restrict__ Kb,
                              const int* __restrict__ pos_ids) {
  const int idx = blockIdx.x;              // ((b*L + pos)*KVH + kh)
  const int rest = idx / KVH_;
  const int posi = rest % L_;
  const int b = rest / L_;
  const int d = threadIdx.x;               // 0..63
  float* k = Kb + (size_t)idx * DH_;
  const float p = (float)pos_ids[b * L_ + posi];
  const float inv_ts = __expf(-(float)d * 0.14391156831212787f); // ln(1e4)/64
  float s, c;
  __sincosf(p * inv_ts, &s, &c);
  const float x1 = k[d], x2 = k[d + 64];
  k[d]      = x1 * c - x2 * s;
  k[d + 64] = x2 * c + x1 * s;
}

__global__ void rope_q_kernel(float* __restrict__ Qe,
                              const int* __restrict__ pos_ids) {
  const int idx = blockIdx.x;              // ((b*LE + q)*H + h)
  const int rest = idx / H_;
  const int q = rest % LE_;
  const int b = rest / LE_;
  const int d = threadIdx.x;
  float* qp = Qe + (size_t)idx * DH_;
  const float p = (float)pos_ids[b * L_ + LV_ + q];
  const float inv_ts = __expf(-(float)d * 0.14391156831212787f);
  float s, c;
  __sincosf(p * inv_ts, &s, &c);
  const float x1 = qp[d], x2 = qp[d + 64];
  qp[d]      = x1 * c - x2 * s;
  qp[d + 64] = x2 * c + x1 * s;
}

// ---------------- attention scores: S = scale * Q.K^T, masked --------
__global__ void attn_scores_kernel(const float* __restrict__ Qe,   // (B,LE,H,DH)
                                   const float* __restrict__ Kb,   // (B,L,KVH,DH)
                                   const unsigned char* __restrict__ mask, // (B,L,L)
                                   float* __restrict__ S) {        // (B,H,LE,L)
  const int bh = blockIdx.y;
  const int b = bh / H_, h = bh - b * H_;
  const int kh = h / G_;
  const int lane = threadIdx.x & 31;
  const int wave = blockIdx.x * (blockDim.x >> 5) + (threadIdx.x >> 5);
  const int ntile = L_ / 16;               // 68
  const int tm = wave / ntile;
  const int tn = wave - tm * ntile;
  if (tm >= LE_ / 16) return;
  const int r = lane & 15;
  const int g = lane >> 4;
  const float* Qrow = Qe + (((size_t)b * LE_ + tm * 16 + r) * H_ + h) * DH_;
  const float* Krow = Kb + (((size_t)b * L_ + tn * 16 + r) * KVH_ + kh) * DH_;
  v8f acc = {};
  for (int k0 = 0; k0 < DH_; k0 += 4) {
    v2f a = *(const v2f*)(Qrow + k0 + 2 * g);
    v2f bf = *(const v2f*)(Krow + k0 + 2 * g);
    acc = wmma4_f32(a, bf, acc);
  }
  const float scale = 0.08838834764831845f;     // 1/sqrt(128)
  const int kpos = tn * 16 + r;
#pragma unroll
  for (int v = 0; v < 8; ++v) {
    const int q = tm * 16 + v + 8 * g;
    float val = acc[v] * scale;
    const unsigned char m = mask[((size_t)b * L_ + (LV_ + q)) * L_ + kpos];
    if (!m) val = -3.4028234663852886e38f;
    S[(((size_t)b * H_ + h) * LE_ + q) * L_ + kpos] = val;
  }
}

// ---------------- softmax over last dim, one wave per row ------------
__global__ void softmax_rows(float* __restrict__ S, int rows, int cols) {
  const int row = blockIdx.x * (blockDim.x >> 5) + (threadIdx.x >> 5);
  const int lane = threadIdx.x & 31;
  if (row >= rows) return;
  float* p = S + (size_t)row * cols;
  float m = -3.4028234663852886e38f;
  for (int c = lane; c < cols; c += 32) m = fmaxf(m, p[c]);
  for (int off = 16; off > 0; off >>= 1) m = fmaxf(m, __shfl_xor(m, off, 32));
  float s = 0.f;
  for (int c = lane; c < cols; c += 32) {
    float e = __expf(p[c] - m);
    p[c] = e;
    s += e;
  }
  for (int off = 16; off > 0; off >>= 1) s += __shfl_xor(s, off, 32);
  const float inv = 1.0f / s;
  for (int c = lane; c < cols; c += 32) p[c] *= inv;
}

// ---------------- O = P @ V (GQA) ------------------------------------
__global__ void attn_pv_kernel(const float* __restrict__ S,   // (B,H,LE,L) probs
                               const float* __restrict__ Vb,  // (B,L,KVH,DH)
                               float* __restrict__ O) {       // (B,LE,H,DH)
  const int bh = blockIdx.y;
  const int b = bh / H_, h = bh - b * H_;
  const int kh = h / G_;
  const int lane = threadIdx.x & 31;
  const int wave = blockIdx.x * (blockDim.x >> 5) + (threadIdx.x >> 5);
  const int ntile = DH_ / 16;              // 8
  const int tm = wave / ntile;
  const int tn = wave - tm * ntile;
  if (tm >= LE_ / 16) return;
  const int r = lane & 15;
  const int g = lane >> 4;
  const float* Prow = S + (((size_t)b * H_ + h) * LE_ + tm * 16 + r) * L_;
  const float* Vp = Vb + ((size_t)b * L_ * KVH_ + kh) * DH_ + tn * 16 + r;
  v8f acc = {};
  for (int k0 = 0; k0 < L_; k0 += 4) {
    v2f a = *(const v2f*)(Prow + k0 + 2 * g);
    v2f bf;
    bf.x = Vp[(size_t)(k0 + 2 * g + 0) * KVDIM];
    bf.y = Vp[(size_t)(k0 + 2 * g + 1) * KVDIM];
    acc = wmma4_f32(a, bf, acc);
  }
#pragma unroll
  for (int v = 0; v < 8; ++v) {
    const int q = tm * 16 + v + 8 * g;
    O[(((size_t)b * LE_ + q) * H_ + h) * DH_ + tn * 16 + r] = acc[v];
  }
}

// ---------------- host side -----------------------------------------
static inline void launch_gemm(const float* A, const float* W, float* C,
                               int M, int N, int K, hipStream_t s) {
  const int tiles = (M / 16) * (N / 16);
  const int blocks = (tiles + 7) / 8;
  gemm_wmma_f32<<<blocks, 256, 0, s>>>(A, W, C, M, N, K);
}

extern "C" void kernel_launch(void* const* d_in, const int* in_sizes, int n_in,
                              void* d_out, int out_size, void* d_ws, size_t ws_size,
                              hipStream_t stream) {
  const float* vlm      = (const float*)d_in[0];   // (B,LV,DV)
  const float* expert   = (const float*)d_in[1];   // (B,LE,DE)
  const int* pos        = (const int*)d_in[2];     // (B,L)
  const unsigned char* mask = (const unsigned char*)d_in[3]; // (B,L,L) bool
  const float* w_ln_vlm = (const float*)d_in[4];
  // d_in[5] Wq_vlm, d_in[8] Wo_vlm: dead — only out_exp is returned
  const float* Wk_vlm   = (const float*)d_in[6];
  const float* Wv_vlm   = (const float*)d_in[7];
  const float* w_ln_exp = (const float*)d_in[9];
  const float* Wq_exp   = (const float*)d_in[10];
  const float* Wk_exp   = (const float*)d_in[11];
  const float* Wv_exp   = (const float*)d_in[12];
  const float* Wo_exp   = (const float*)d_in[13];
  float* out = (float*)d_out;

  // workspace layout (floats); total ~15.5M floats (~62 MB)
  float* ws = (float*)d_ws;
  float* hv = ws;                                  // B*LV*DV
  float* he = hv + (size_t)B_ * LV_ * DV_;         // B*LE*DE
  float* Kb = he + (size_t)B_ * LE_ * DE_;         // B*L*KVDIM
  float* Vb = Kb + (size_t)B_ * L_ * KVDIM;        // B*L*KVDIM
  float* Qe = Vb + (size_t)B_ * L_ * KVDIM;        // B*LE*QDIM
  float* S  = Qe + (size_t)B_ * LE_ * QDIM;        // B*H*LE*L
  float* O  = S  + (size_t)B_ * H_ * LE_ * L_;     // B*LE*QDIM

  // 1) RMS norms
  rmsnorm_kernel<<<B_ * LV_, 256, 0, stream>>>(vlm, w_ln_vlm, hv, DV_);
  rmsnorm_kernel<<<B_ * LE_, 256, 0, stream>>>(expert, w_ln_exp, he, DE_);

  // 2) projections
  for (int b = 0; b < B_; ++b) {
    const float* hvb = hv + (size_t)b * LV_ * DV_;
    const float* heb = he + (size_t)b * LE_ * DE_;
    float* Kv = Kb + (size_t)b * L_ * KVDIM;
    float* Vv = Vb + (size_t)b * L_ * KVDIM;
    // hot GEMMs: LDS-tiled + async double buffering
    dim3 gt(KVDIM / BN, LV_ / BM);
    gemm_tiled_async<<<gt, 256, 0, stream>>>(hvb, Wk_vlm, Kv, LV_, KVDIM, DV_);
    gemm_tiled_async<<<gt, 256, 0, stream>>>(hvb, Wv_vlm, Vv, LV_, KVDIM, DV_);
    // small expert GEMMs (M=64)
    launch_gemm(heb, Wk_exp, Kv + (size_t)LV_ * KVDIM, LE_, KVDIM, DE_, stream);
    launch_gemm(heb, Wv_exp, Vv + (size_t)LV_ * KVDIM, LE_, KVDIM, DE_, stream);
    launch_gemm(heb, Wq_exp, Qe + (size_t)b * LE_ * QDIM, LE_, QDIM, DE_, stream);
  }

  // 3) RoPE on K (all positions) and expert Q
  rope_k_kernel<<<B_ * L_ * KVH_, 64, 0, stream>>>(Kb, pos);
  rope_q_kernel<<<B_ * LE_ * H_, 64, 0, stream>>>(Qe, pos);

  // 4) scores (scale + mask)
  {
    dim3 grid((4 * (L_ / 16) + 7) / 8, B_ * H_);
    attn_scores_kernel<<<grid, 256, 0, stream>>>(Qe, Kb, mask, S);
  }

  // 5) softmax over 1088 keys, one wave per (b,h,q) row
  softmax_rows<<<(B_ * H_ * LE_ + 7) / 8, 256, 0, stream>>>(S, B_ * H_ * LE_, L_);

  // 6) P @ V
  {
    dim3 grid((4 * (DH_ / 16) + 7) / 8, B_ * H_);
    attn_pv_kernel<<<grid, 256, 0, stream>>>(S, Vb, O);
  }

  // 7) out_exp = expert_embeds + O @ Wo_exp
  {
    const int tiles = ((B_ * LE_) / 16) * (DE_ / 16);
    out_proj_wmma<<<(tiles + 7) / 8, 256, 0, stream>>>(O, Wo_exp, expert, out,
                                                       B_ * LE_, DE_, QDIM);
  }
}